// ProbAttention_23656679867656
// MI455X (gfx1250) — compile-verified
//
#include <hip/hip_runtime.h>
#include <hip/hip_bf16.h>

typedef __attribute__((ext_vector_type(2))) float v2f;
typedef __attribute__((ext_vector_type(8))) float v8f;

__device__ __forceinline__ v8f wmma_f32_4(v2f a, v2f b, v8f c) {
    // V_WMMA_F32_16X16X4_F32 : D(16x16,f32) = A(16x4,f32) x B(4x16,f32) + C
    return __builtin_amdgcn_wmma_f32_16x16x4_f32(
        /*neg_a=*/false, a, /*neg_b=*/false, b,
        /*c_mod=*/(short)0, c, /*reuse_a=*/false, /*reuse_b=*/false);
}

#define L 1024
#define H 8
#define D 64
#define DSK 256
#define NQ 256

// -------------------------------------------------------------------------
// Kernel 1: per (bh, q-tile of 16): compute 16x1024 score block with WMMA
// into LDS, then gather sampled columns -> measure = max - sum/1024.
// -------------------------------------------------------------------------
__global__ __launch_bounds__(256) void k_measure(const float* __restrict__ q,
                                                 const float* __restrict__ kmat,
                                                 const int* __restrict__ index_key,
                                                 float* __restrict__ measure) {
    const int qt = blockIdx.x;     // 0..63
    const int bh = blockIdx.y;     // 0..31
    const int b = bh >> 3, h = bh & 7;
    const int q0 = qt * 16;

    __shared__ float sQ[16][64];
    __shared__ float sS[16][1024];
    __shared__ float redm[16][16];
    __shared__ float reds[16][16];

    const int t = threadIdx.x;
    for (int i = t; i < 16 * 64; i += 256) {
        int r = i >> 6, d = i & 63;
        sQ[r][d] = q[(((size_t)b * L + (q0 + r)) * H + h) * D + d];
    }
    __syncthreads();

    const int wave = t >> 5, lane = t & 31;
    const int m = lane & 15, half = lane >> 4;
    for (int kt = wave; kt < 64; kt += 8) {
        const int k0 = kt * 16;
        v8f acc = {0.f, 0.f, 0.f, 0.f, 0.f, 0.f, 0.f, 0.f};
        const float* krow = &kmat[(((size_t)b * L + (k0 + m)) * H + h) * D];
        #pragma unroll
        for (int c = 0; c < 64; c += 4) {
            v2f a, bb;
            a.x = sQ[m][c + 2 * half];       // A: lanes0-15 K=c,c+1 ; lanes16-31 K=c+2,c+3
            a.y = sQ[m][c + 2 * half + 1];
            bb.x = krow[c + half];           // B: VGPR0 rows K=c(+half)
            bb.y = krow[c + 2 + half];       //    VGPR1 rows K=c+2(+half)
            acc = wmma_f32_4(a, bb, acc);
        }
        #pragma unroll
        for (int r = 0; r < 8; r++) sS[r + 8 * half][k0 + m] = acc[r];
    }
    __syncthreads();

    // sampled gather: 16 threads per row, 16 samples each
    const int row = t >> 4, seg = t & 15;
    const int qg = q0 + row;
    float mx = -3.0e38f, sm = 0.f;
    const int* idxp = &index_key[(size_t)qg * DSK + seg * 16];
    #pragma unroll
    for (int j = 0; j < 16; j++) {
        float s = sS[row][idxp[j]];
        mx = fmaxf(mx, s);
        sm += s;
    }
    redm[row][seg] = mx;
    reds[row][seg] = sm;
    __syncthreads();
    if (seg == 0) {
        float M = -3.0e38f, S = 0.f;
        #pragma unroll
        for (int j = 0; j < 16; j++) { M = fmaxf(M, redm[row][j]); S += reds[row][j]; }
        measure[(size_t)bh * L + qg] = M - S * (1.0f / 1024.0f);
    }
}

// -------------------------------------------------------------------------
// Kernel 2: rank-based top-256 per (b,h). Tie-break (==, lower index first)
// matches jax.lax.top_k ordering; ranks are unique so qidx fully written.
// -------------------------------------------------------------------------
__global__ __launch_bounds__(256) void k_topk(const float* __restrict__ measure,
                                              int* __restrict__ qidx) {
    const int bh = blockIdx.x;
    __shared__ float mm[1024];
    const int t = threadIdx.x;
    for (int i = t; i < L; i += 256) mm[i] = measure[(size_t)bh * L + i];
    __syncthreads();
    for (int qi = t; qi < L; qi += 256) {
        const float mv = mm[qi];
        int rank = 0;
        for (int j = 0; j < L; j++) {
            float mj = mm[j];
            rank += (int)((mj > mv) || (mj == mv && j < qi));
        }
        if (rank < NQ) qidx[(size_t)bh * NQ + rank] = qi;
    }
}

// -------------------------------------------------------------------------
// Kernel 3a: out[b,q,h,:] = mean_k V[b,k,h,:]  (for all q rows)
// -------------------------------------------------------------------------
__global__ __launch_bounds__(256) void k_init(const float* __restrict__ v,
                                              float* __restrict__ out) {
    const int bh = blockIdx.x;
    const int b = bh >> 3, h = bh & 7;
    __shared__ float part[4][64];
    __shared__ float meanv[64];
    const int t = threadIdx.x;
    const int d = t & 63, sl = t >> 6;
    float acc = 0.f;
    for (int k = sl; k < L; k += 4) acc += v[(((size_t)b * L + k) * H + h) * D + d];
    part[sl][d] = acc;
    __syncthreads();
    if (t < 64) meanv[t] = (part[0][t] + part[1][t] + part[2][t] + part[3][t]) * (1.0f / 1024.0f);
    __syncthreads();
    for (int i = t; i < L * D; i += 256) {
        int qq = i >> 6, dd = i & 63;
        out[(((size_t)b * L + qq) * H + h) * D + dd] = meanv[dd];
    }
}

// -------------------------------------------------------------------------
// Kernel 3b: per (bh, tile of 16 selected rows): WMMA scores -> softmax in
// LDS -> WMMA P.V -> scatter into out at the selected query rows.
// -------------------------------------------------------------------------
__global__ __launch_bounds__(256) void k_attn(const float* __restrict__ q,
                                              const float* __restrict__ kmat,
                                              const float* __restrict__ v,
                                              const int* __restrict__ qidx,
                                              float* __restrict__ out) {
    const int st = blockIdx.x;     // 0..15
    const int bh = blockIdx.y;     // 0..31
    const int b = bh >> 3, h = bh & 7;

    __shared__ int qsel[16];
    __shared__ float sQ[16][64];   // Q tile; reused as ctx accumulator for P.V
    __shared__ float sS[16][1024];
    __shared__ float red[16][16];
    __shared__ float rowmax[16];
    __shared__ float rowsum[16];

    const int t = threadIdx.x;
    if (t < 16) qsel[t] = qidx[(size_t)bh * NQ + st * 16 + t];
    __syncthreads();
    for (int i = t; i < 16 * 64; i += 256) {
        int r = i >> 6, d = i & 63;
        sQ[r][d] = q[(((size_t)b * L + qsel[r]) * H + h) * D + d];
    }
    __syncthreads();

    const int wave = t >> 5, lane = t & 31;
    const int m = lane & 15, half = lane >> 4;

    // ---- scores: S(16x1024) = Q_sel . K^T via WMMA ----
    for (int kt = wave; kt < 64; kt += 8) {
        const int k0 = kt * 16;
        v8f acc = {0.f, 0.f, 0.f, 0.f, 0.f, 0.f, 0.f, 0.f};
        const float* krow = &kmat[(((size_t)b * L + (k0 + m)) * H + h) * D];
        #pragma unroll
        for (int c = 0; c < 64; c += 4) {
            v2f a, bb;
            a.x = sQ[m][c + 2 * half];
            a.y = sQ[m][c + 2 * half + 1];
            bb.x = krow[c + half];
            bb.y = krow[c + 2 + half];
            acc = wmma_f32_4(a, bb, acc);
        }
        #pragma unroll
        for (int r = 0; r < 8; r++) sS[r + 8 * half][k0 + m] = acc[r];
    }
    __syncthreads();

    // ---- softmax over each row of sS (scale then exp; fold 1/sum later) ----
    const float scale = 0.125f;  // 1/sqrt(64)
    const int row = t >> 4, seg = t & 15;
    float mx = -3.0e38f;
    for (int c = seg * 64; c < seg * 64 + 64; c++) mx = fmaxf(mx, sS[row][c] * scale);
    red[row][seg] = mx;
    __syncthreads();
    if (seg == 0) {
        float M = -3.0e38f;
        #pragma unroll
        for (int j = 0; j < 16; j++) M = fmaxf(M, red[row][j]);
        rowmax[row] = M;
    }
    __syncthreads();
    float sm = 0.f;
    for (int c = seg * 64; c < seg * 64 + 64; c++) {
        float e = __expf(sS[row][c] * scale - rowmax[row]);
        sS[row][c] = e;
        sm += e;
    }
    red[row][seg] = sm;
    __syncthreads();
    if (seg == 0) {
        float S = 0.f;
        #pragma unroll
        for (int j = 0; j < 16; j++) S += red[row][j];
        rowsum[row] = S;
    }
    __syncthreads();   // sQ dead from here: reuse as ctx accumulator

    // ---- ctx(16x64) = P(16x1024) . V(1024x64) via WMMA ----
    // 8 waves: wave = (khalf<<2) | dtile. Each wave: one 16x16 d-tile over a
    // 512-long K half-range (128 WMMAs). Partials combined in LDS (sQ).
    {
        const int dt = wave & 3;          // d-tile 0..3
        const int d0 = dt * 16;
        const int khalf = wave >> 2;      // 0 or 1
        const int kbase = khalf * 512;
        v8f acc = {0.f, 0.f, 0.f, 0.f, 0.f, 0.f, 0.f, 0.f};
        const float* vbase = &v[(((size_t)b * L) * H + h) * D];  // +k*H*D + d
        for (int c = kbase; c < kbase + 512; c += 4) {
            v2f a, bb;
            a.x = sS[m][c + 2 * half];                 // A = P row m, K cols c+2*half..+1
            a.y = sS[m][c + 2 * half + 1];
            bb.x = vbase[(size_t)(c + half) * H * D + d0 + m];      // B row K=c+half
            bb.y = vbase[(size_t)(c + 2 + half) * H * D + d0 + m];  // B row K=c+2+half
            acc = wmma_f32_4(a, bb, acc);
        }
        if (khalf == 0) {
            #pragma unroll
            for (int r = 0; r < 8; r++) sQ[r + 8 * half][d0 + m] = acc[r];
        }
        __syncthreads();
        if (khalf == 1) {
            #pragma unroll
            for (int r = 0; r < 8; r++) sQ[r + 8 * half][d0 + m] += acc[r];
        }
        __syncthreads();
    }

    // ---- normalize and scatter to selected query rows ----
    for (int o = t; o < 16 * 64; o += 256) {
        const int r = o >> 6, dd = o & 63;
        out[(((size_t)b * L + qsel[r]) * H + h) * D + dd] = sQ[r][dd] / rowsum[r];
    }
}

extern "C" void kernel_launch(void* const* d_in, const int* in_sizes, int n_in,
                              void* d_out, int out_size, void* d_ws, size_t ws_size,
                              hipStream_t stream) {
    const float* q = (const float*)d_in[0];
    const float* k = (const float*)d_in[1];
    const float* v = (const float*)d_in[2];
    const int* index_key = (const int*)d_in[3];
    float* out = (float*)d_out;

    float* measure = (float*)d_ws;                                        // 32*1024 f32 = 128KB
    int* qidx = (int*)((char*)d_ws + (size_t)32 * 1024 * sizeof(float));  // 32*256 i32 = 32KB

    k_measure<<<dim3(64, 32), 256, 0, stream>>>(q, k, index_key, measure);
    k_topk<<<dim3(32), 256, 0, stream>>>(measure, qidx);
    k_init<<<dim3(32), 256, 0, stream>>>(v, out);
    k_attn<<<dim3(16, 32), 256, 0, stream>>>(q, k, v, qidx, out);
}